// QuantumLayer_60988535603889
// MI455X (gfx1250) — compile-verified
//
#include <hip/hip_runtime.h>
#include <math.h>

// out[n][w] = prod_{j<=w} cos(pi * x[n][j])   (analytic collapse of the circuit)
// Prefix products as a triangular GEMM in the log2 domain:
//   magnitude: exp2( log2|c| * Tri^T )  via V_WMMA_F32_16X16X4_F32 (4 K-steps)
//   sign:      parity( 1[c<0]  * Tri^T ) via a second WMMA accumulator
// Raw HW transcendentals: v_cos_f32 computes cos(2*pi*s) -> cos(pi*x) = v_cos(0.5*x);
// v_log_f32 = log2 (log2(0) = -inf), v_exp_f32 = exp2 (exp2(-inf) = 0).
// |cos(pi*x)| is never denormal for fp32 x (>= pi*2^-25 when nonzero), so the
// libdevice denorm pre-scaling is unnecessary.

typedef __attribute__((ext_vector_type(2))) float v2f;
typedef __attribute__((ext_vector_type(8))) float v8f;

__global__ __launch_bounds__(256) void qprefix_wmma(const float* __restrict__ x,
                                                    float* __restrict__ out) {
    const int lane  = threadIdx.x & 31;
    const int wave  = threadIdx.x >> 5;
    const int tile  = blockIdx.x * 8 + wave;   // 16 samples per tile, 32 tiles total
    const int m     = lane & 15;               // A: row M ; B,C,D: column N
    const int khalf = lane >> 4;               // which K-pair of the 4-wide slice

    const float* xrow = x + (size_t)(tile * 16 + m) * 16;

    v8f accL = {0.f,0.f,0.f,0.f,0.f,0.f,0.f,0.f};   // prefix sum of log2|cos|
    v8f accS = {0.f,0.f,0.f,0.f,0.f,0.f,0.f,0.f};   // prefix count of negatives

#pragma unroll
    for (int kk = 0; kk < 4; ++kk) {
        const int k0 = kk * 4 + khalf * 2;     // global K index of this lane's pair

        // A-matrix slice (16x4 f32): lane holds (M=m, K=k0) and (M=m, K=k0+1)
        const float c0 = __builtin_amdgcn_cosf(xrow[k0]     * 0.5f);  // cos(pi*x)
        const float c1 = __builtin_amdgcn_cosf(xrow[k0 + 1] * 0.5f);

        v2f aL, aS, b;
        aL.x = __builtin_amdgcn_logf(__builtin_fabsf(c0));  // v_log_f32, |.| folded
        aL.y = __builtin_amdgcn_logf(__builtin_fabsf(c1));
        aS.x = (c0 < 0.0f) ? 1.0f : 0.0f;
        aS.y = (c1 < 0.0f) ? 1.0f : 0.0f;

        // B-matrix slice (4x16 f32): lower-triangular-ones^T: B[k][n] = (k <= n)
        // lane holds (K=k0, N=m) and (K=k0+1, N=m)
        b.x = (k0     <= m) ? 1.0f : 0.0f;
        b.y = (k0 + 1 <= m) ? 1.0f : 0.0f;

        // 8 args: (neg_a, A, neg_b, B, c_mod, C, reuse_a, reuse_b)
        accL = __builtin_amdgcn_wmma_f32_16x16x4_f32(false, aL, false, b,
                                                     (short)0, accL, false, false);
        accS = __builtin_amdgcn_wmma_f32_16x16x4_f32(false, aS, false, b,
                                                     (short)0, accS, false, false);
    }

    // D layout: VGPR i, lanes 0-15 -> (M=i, N=lane); lanes 16-31 -> (M=i+8, N=lane-16)
#pragma unroll
    for (int i = 0; i < 8; ++i) {
        const int row = i + khalf * 8;                     // sample within tile
        const float mag = __builtin_amdgcn_exp2f(accL[i]); // v_exp_f32; exp2(-inf)=0
        const unsigned cnt = (unsigned)(int)accS[i];       // exact small integer
        // flip sign bit iff odd count of negative factors
        const unsigned bits = __float_as_uint(mag) ^ (cnt << 31);
        out[(size_t)(tile * 16 + row) * 16 + m] = __uint_as_float(bits);
    }
}

extern "C" void kernel_launch(void* const* d_in, const int* in_sizes, int n_in,
                              void* d_out, int out_size, void* d_ws, size_t ws_size,
                              hipStream_t stream) {
    (void)in_sizes; (void)n_in; (void)d_ws; (void)ws_size; (void)out_size;
    const float* x = (const float*)d_in[0];  // (512, 16) float32
    float* out = (float*)d_out;              // (512, 16) float32
    // 512 samples / 16 per wave-tile = 32 waves = 4 blocks x 8 waves (256 threads)
    qprefix_wmma<<<4, 256, 0, stream>>>(x, out);
}